// FusedMoEGate_33981781246239
// MI455X (gfx1250) — compile-verified
//
#include <hip/hip_runtime.h>
#include <math.h>

#define D_X        1024
#define D_TEXT     768
#define FUSION_DIM 1792
#define BATCH      256
#define HW         1024
#define NEXP       16

typedef __attribute__((ext_vector_type(2))) float v2f;
typedef __attribute__((ext_vector_type(8))) float v8f;

// ---------------------------------------------------------------------------
// 1) Mean-pool x over H*W. One wave32 per (b,c) channel (4 KB contiguous).
//    8 x float4 loads per lane -> global_load_b128, then xor-shuffle reduce.
// ---------------------------------------------------------------------------
__global__ void pool_kernel(const float* __restrict__ x, float* __restrict__ fused) {
    const int wave = (blockIdx.x * blockDim.x + threadIdx.x) >> 5; // = b*D_X + c
    const int lane = threadIdx.x & 31;
    const float* base = x + (size_t)wave * HW;

    float s = 0.0f;
#pragma unroll
    for (int i = 0; i < 8; ++i) {
        float4 v = *(const float4*)(base + i * 128 + lane * 4);
        s += v.x + v.y + v.z + v.w;
    }
#pragma unroll
    for (int m = 16; m >= 1; m >>= 1) s += __shfl_xor(s, m, 32);

    if (lane == 0) {
        const int b = wave / D_X;
        const int c = wave - b * D_X;
        fused[(size_t)b * FUSION_DIM + c] = s * (1.0f / 1024.0f);
    }
}

// ---------------------------------------------------------------------------
// 2) Copy degraded into fused[:, 1024:1792]
// ---------------------------------------------------------------------------
__global__ void concat_kernel(const float* __restrict__ degraded, float* __restrict__ fused) {
    const int i = blockIdx.x * blockDim.x + threadIdx.x;
    if (i < BATCH * D_TEXT) {
        const int b = i / D_TEXT;
        const int j = i - b * D_TEXT;
        fused[(size_t)b * FUSION_DIM + D_X + j] = degraded[i];
    }
}

__device__ __forceinline__ float gelu_exact(float v) {
    return 0.5f * v * (1.0f + erff(v * 0.70710678118654752f));
}

// ---------------------------------------------------------------------------
// 3) fused(256x1792) @ w_fusion(1792x1792) + bias, exact GELU.
//    fp32 WMMA (V_WMMA_F32_16X16X4_F32), exact numerics, K stepped by 4.
//    Block = 2 waves; each wave owns four 16x16 output tiles (A reused 4x).
//    Grid = (16 row bands, 14 column groups of 128).
// ---------------------------------------------------------------------------
__global__ void fusion_gemm_kernel(const float* __restrict__ A,
                                   const float* __restrict__ W,
                                   const float* __restrict__ bias,
                                   float* __restrict__ act) {
    const int lane = threadIdx.x & 31;
    const int wav  = threadIdx.x >> 5;
    const int row0 = blockIdx.x * 16;
    const int n0   = (blockIdx.y * 8 + wav * 4) * 16;  // first of 4 N-tiles

    const int m    = lane & 15;        // A row within band / C column index
    const int half = lane >> 4;        // 0: K 0..1 | 1: K 2..3 (per ISA layout)
    const int koff = half * 2;
    const int ncol = lane & 15;

    v8f acc0 = {}, acc1 = {}, acc2 = {}, acc3 = {};
    const float* arow = A + (size_t)(row0 + m) * FUSION_DIM;

    for (int k = 0; k < FUSION_DIM; k += 4) {
        v2f a = *(const v2f*)(arow + k + koff);            // A[m][k+koff .. +1]
        const float* w0 = W + (size_t)(k + koff) * FUSION_DIM;
        const float* w1 = W + (size_t)(k + koff + 1) * FUSION_DIM;
        v2f b0 = { w0[n0 +  0 + ncol], w1[n0 +  0 + ncol] };
        v2f b1 = { w0[n0 + 16 + ncol], w1[n0 + 16 + ncol] };
        v2f b2 = { w0[n0 + 32 + ncol], w1[n0 + 32 + ncol] };
        v2f b3 = { w0[n0 + 48 + ncol], w1[n0 + 48 + ncol] };
        acc0 = __builtin_amdgcn_wmma_f32_16x16x4_f32(false, a, false, b0, (short)0, acc0, false, false);
        acc1 = __builtin_amdgcn_wmma_f32_16x16x4_f32(false, a, false, b1, (short)0, acc1, false, false);
        acc2 = __builtin_amdgcn_wmma_f32_16x16x4_f32(false, a, false, b2, (short)0, acc2, false, false);
        acc3 = __builtin_amdgcn_wmma_f32_16x16x4_f32(false, a, false, b3, (short)0, acc3, false, false);
    }

    // Epilogue: C/D layout -> VGPR v holds M = v (lanes 0-15) / v+8 (lanes 16-31)
#pragma unroll
    for (int v = 0; v < 8; ++v) {
        const int row = row0 + v + half * 8;
        float* arow_out = act + (size_t)row * FUSION_DIM;
        {
            int col = n0 + 0 + ncol;
            arow_out[col] = gelu_exact(acc0[v] + bias[col]);
        }
        {
            int col = n0 + 16 + ncol;
            arow_out[col] = gelu_exact(acc1[v] + bias[col]);
        }
        {
            int col = n0 + 32 + ncol;
            arow_out[col] = gelu_exact(acc2[v] + bias[col]);
        }
        {
            int col = n0 + 48 + ncol;
            arow_out[col] = gelu_exact(acc3[v] + bias[col]);
        }
    }
}

// ---------------------------------------------------------------------------
// 4) Gate logits (1792 -> 16), softmax probs, top-2 gates + indices.
//    One block per batch row; 16x16 partial-dot then serial 16-wide epilogue.
// ---------------------------------------------------------------------------
__global__ void gate_kernel(const float* __restrict__ act,
                            const float* __restrict__ w_gate,
                            float* __restrict__ out) {
    __shared__ float part[16][17];
    __shared__ float logits[16];

    const int b = blockIdx.x;
    const int t = threadIdx.x;
    const int n = t & 15;
    const int chunk = t >> 4;   // 0..15, each covers 112 K values

    const float* arow = act + (size_t)b * FUSION_DIM;
    float s = 0.0f;
    const int k0 = chunk * 112;
    for (int k = k0; k < k0 + 112; ++k)
        s += arow[k] * w_gate[k * NEXP + n];
    part[chunk][n] = s;
    __syncthreads();

    if (t < 16) {
        float tot = 0.0f;
#pragma unroll
        for (int c = 0; c < 16; ++c) tot += part[c][t];
        logits[t] = tot;
    }
    __syncthreads();

    if (t == 0) {
        float* gates = out;                            // [0, 4096)
        float* probs = out + BATCH * NEXP + 1;         // [4097, 8193)
        float* tidx  = out + 2 * BATCH * NEXP + 1;     // [8193, 8705)

        // full softmax (probs_for_mi)
        float mx = logits[0];
#pragma unroll
        for (int i = 1; i < 16; ++i) mx = fmaxf(mx, logits[i]);
        float e[16];
        float denom = 0.0f;
#pragma unroll
        for (int i = 0; i < 16; ++i) { e[i] = expf(logits[i] - mx); denom += e[i]; }
        const float inv = 1.0f / denom;
#pragma unroll
        for (int i = 0; i < 16; ++i) probs[b * NEXP + i] = e[i] * inv;

        // top-2 (ties -> lowest index, matching lax.top_k)
        int i0 = 0;
        for (int i = 1; i < 16; ++i) if (logits[i] > logits[i0]) i0 = i;
        int i1 = (i0 == 0) ? 1 : 0;
        for (int i = 0; i < 16; ++i) if (i != i0 && logits[i] > logits[i1]) i1 = i;

        const float l0 = logits[i0], l1 = logits[i1];
        const float g0 = 1.0f / (1.0f + expf(l1 - l0));   // stable: l1 <= l0
        const float g1 = 1.0f - g0;

#pragma unroll
        for (int i = 0; i < 16; ++i) gates[b * NEXP + i] = 0.0f;
        gates[b * NEXP + i0] = g0;
        gates[b * NEXP + i1] = g1;
        tidx[b * 2 + 0] = (float)i0;
        tidx[b * 2 + 1] = (float)i1;

        if (b == 0) out[BATCH * NEXP] = 0.0f;          // moe_loss
    }
}

// ---------------------------------------------------------------------------
extern "C" void kernel_launch(void* const* d_in, const int* in_sizes, int n_in,
                              void* d_out, int out_size, void* d_ws, size_t ws_size,
                              hipStream_t stream) {
    (void)in_sizes; (void)n_in; (void)out_size; (void)ws_size;
    const float* x        = (const float*)d_in[0];
    const float* degraded = (const float*)d_in[1];
    const float* w_fusion = (const float*)d_in[2];
    const float* b_fusion = (const float*)d_in[3];
    const float* w_gate   = (const float*)d_in[4];

    float* fused = (float*)d_ws;                               // 256*1792 f32
    float* act   = fused + (size_t)BATCH * FUSION_DIM;         // 256*1792 f32
    float* out   = (float*)d_out;

    // 262144 channels, 8 waves (one channel each) per 256-thread block
    pool_kernel<<<(BATCH * D_X) / 8, 256, 0, stream>>>(x, fused);
    concat_kernel<<<(BATCH * D_TEXT + 255) / 256, 256, 0, stream>>>(degraded, fused);

    dim3 g(16, 14);   // 16 row bands x (14 * 128 columns) = 256 x 1792
    fusion_gemm_kernel<<<g, 64, 0, stream>>>(fused, w_fusion, b_fusion, act);

    gate_kernel<<<BATCH, 256, 0, stream>>>(act, w_gate, out);
}